// BoundleAdjustment_36094905155781
// MI455X (gfx1250) — compile-verified
//
#include <hip/hip_runtime.h>
#include <stdint.h>

// ---------------------------------------------------------------------------
// Bundle-adjustment residual for MI455X (gfx1250).
//
// Strategy: the 4096x7 pose table (112 KB) is staged once per workgroup into
// LDS (320 KB/WGP on CDNA5 -> 2 workgroups/WGP still fit) using the Tensor
// Data Mover (TENSORcnt-tracked DMA). Random per-observation pose gathers
// then hit LDS; the 3 MB patch table stays L2-resident (192 MB L2). All other
// traffic is streamed/coalesced. Transcendentals (atan2/sqrt) dominate ALU.
// ---------------------------------------------------------------------------

#define BLOCK_C     256
#define OBS_PER_TH  8
#define OBS_PER_BLK (BLOCK_C * OBS_PER_TH)

#ifndef __has_builtin
#define __has_builtin(x) 0
#endif

#if __has_builtin(__builtin_amdgcn_tensor_load_to_lds) && __has_builtin(__builtin_amdgcn_s_wait_tensorcnt)
#define USE_TDM 1
#else
#define USE_TDM 0
#endif

#if __has_builtin(__builtin_amdgcn_global_load_async_to_lds_b128) && __has_builtin(__builtin_amdgcn_s_wait_asynccnt)
#define USE_ASYNC 1
#else
#define USE_ASYNC 0
#endif

typedef unsigned int u32x4 __attribute__((ext_vector_type(4)));
typedef int          i32x8 __attribute__((ext_vector_type(8)));
typedef int          i32x4 __attribute__((ext_vector_type(4)));

__global__ __launch_bounds__(BLOCK_C) void ba_residual_kernel(
    const float* __restrict__ poses,         // (N_POSES, 7) [t.xyz, q.xyzw]
    const float* __restrict__ patch_coords,  // (N_PATCH, 2)
    const float* __restrict__ elev,          // (N_PATCH, 1)
    const int*   __restrict__ poses_idx,     // (N_OBS)
    const int*   __restrict__ patch_idx,     // (N_OBS)
    const float* __restrict__ target,        // (N_OBS, 3)
    const float* __restrict__ weights,       // (N_OBS, 1)
    float*       __restrict__ out,           // (N_OBS, 3)
    int n_obs, int pose_dwords)
{
  extern __shared__ float shPoses[];   // pose_dwords * 4 bytes (112 KB here)
  const int tid = threadIdx.x;

  // ---- Stage full pose table into LDS -------------------------------------
#if USE_TDM
  // Tensor Data Mover: one DMA descriptor moves the whole table.
  // D# per CDNA5 ISA ch.8: 1-row tile of pose_dwords 4-byte elements.
  // (pose_dwords = 28672 fits the 16-bit tile_dim0 field.)
  if (tid == 0) {
    const uint64_t ga = (uint64_t)(uintptr_t)poses;
    const uint32_t la = (uint32_t)(uintptr_t)shPoses;  // low 32 bits = LDS byte addr
    const uint32_t nd = (uint32_t)pose_dwords;
    u32x4 g0;
    g0.x = 1u;                                                 // count=1 (valid user D#)
    g0.y = la;                                                 // lds_addr
    g0.z = (uint32_t)(ga & 0xFFFFFFFFu);                       // global_addr[31:0]
    g0.w = (uint32_t)((ga >> 32) & 0x01FFFFFFu) | (2u << 30);  // global_addr[56:32] | type=2
    i32x8 g1;
    g1[0] = (int)(2u << 16);                            // data_size = 4 bytes
    g1[1] = (int)((nd & 0xFFFFu) << 16);                // tensor_dim0[15:0]
    g1[2] = (int)(((nd >> 16) & 0xFFFFu) | (1u << 16)); // tensor_dim0[31:16] | tensor_dim1 = 1
    g1[3] = (int)((nd & 0xFFFFu) << 16);                // tile_dim0 = nd
    g1[4] = 0;                                          // tile_dim1/2 unused
    g1[5] = (int)nd;                                    // tensor_dim0_stride[31:0]
    g1[6] = (int)((nd & 0xFFFFu) << 16);                // dim0_stride[47:32]=0 | dim1_stride[15:0]
    g1[7] = (int)(nd >> 16);                            // dim1_stride[47:16]
    i32x4 g2 = {1, 1, 0, 0};                            // tensor_dim2=1, tensor_dim3=1
    i32x4 g3 = {0, 0, 0, 0};
    i32x8 g4 = {0, 0, 0, 0, 0, 0, 0, 0};                // extra operand (6-arg form): zero-filled
    __builtin_amdgcn_tensor_load_to_lds(g0, g1, g2, g3, g4, 0);
    __builtin_amdgcn_s_wait_tensorcnt(0);
  }
#elif USE_ASYNC
  {
    const int chunks = pose_dwords >> 2;
    for (int c = tid; c < chunks; c += BLOCK_C) {
      __builtin_amdgcn_global_load_async_to_lds_b128(
          (const __attribute__((address_space(1))) void*)(poses + 4 * c),
          (__attribute__((address_space(3))) void*)(shPoses + 4 * c), 0, 0);
    }
    for (int j = (chunks << 2) + tid; j < pose_dwords; j += BLOCK_C)
      shPoses[j] = poses[j];
    __builtin_amdgcn_s_wait_asynccnt(0);
  }
#else
  {
    const int chunks = pose_dwords >> 2;
    const float4* src4 = reinterpret_cast<const float4*>(poses);
    float4* dst4 = reinterpret_cast<float4*>(shPoses);
    for (int c = tid; c < chunks; c += BLOCK_C) dst4[c] = src4[c];
    for (int j = (chunks << 2) + tid; j < pose_dwords; j += BLOCK_C)
      shPoses[j] = poses[j];
  }
#endif
  __syncthreads();

  // ---- Per-observation residuals ------------------------------------------
  const int base = blockIdx.x * OBS_PER_BLK + tid;
#pragma unroll
  for (int k = 0; k < OBS_PER_TH; ++k) {
    const int o = base + k * BLOCK_C;
    if (o >= n_obs) break;

    if (k + 1 < OBS_PER_TH) {  // speculative prefetch of next iteration's stream
      __builtin_prefetch(&target[3 * (o + BLOCK_C)], 0, 1);
      __builtin_prefetch(&weights[o + BLOCK_C], 0, 1);
    }

    const int pi = poses_idx[o] * 7;
    const int gi = patch_idx[o];

    const float tx = shPoses[pi + 0], ty = shPoses[pi + 1], tz = shPoses[pi + 2];
    const float qx = shPoses[pi + 3], qy = shPoses[pi + 4];
    const float qz = shPoses[pi + 5], qw = shPoses[pi + 6];

    const float px = patch_coords[2 * gi];
    const float py = patch_coords[2 * gi + 1];
    const float pz = elev[gi];

    // Rotation with unnormalized quaternion:
    //   rot = p + (2/|q|^2) * (qw*(qv x p) + qv x (qv x p))
    // (identical to normalize-then-rotate; avoids sqrt + 4 divides)
    const float s  = 2.0f / fmaf(qw, qw, fmaf(qz, qz, fmaf(qy, qy, qx * qx)));
    const float ux = qy * pz - qz * py;
    const float uy = qz * px - qx * pz;
    const float uz = qx * py - qy * px;
    const float wxv = fmaf(qw, ux, qy * uz - qz * uy);
    const float wyv = fmaf(qw, uy, qz * ux - qx * uz);
    const float wzv = fmaf(qw, uz, qx * uy - qy * ux);
    const float X = fmaf(s, wxv, px) + tx;
    const float Y = fmaf(s, wyv, py) + ty;
    const float Z = fmaf(s, wzv, pz) + tz;

    // cart -> polar (reference computes r via rho)
    const float rho = __builtin_sqrtf(fmaf(X, X, Y * Y));
    const float r   = __builtin_sqrtf(fmaf(rho, rho, Z * Z));
    const float az  = atan2f(Y, X);
    const float el  = atan2f(Z, rho);

    const float w  = weights[o];
    const float t0 = target[3 * o + 0];
    const float t1 = target[3 * o + 1];
    const float t2 = target[3 * o + 2];
    out[3 * o + 0] = (r  - t0) * w;
    out[3 * o + 1] = (az - t1) * w;
    out[3 * o + 2] = (el - t2) * w;
  }
}

extern "C" void kernel_launch(void* const* d_in, const int* in_sizes, int n_in,
                              void* d_out, int out_size, void* d_ws, size_t ws_size,
                              hipStream_t stream) {
  const float* poses        = (const float*)d_in[0];
  const float* patch_coords = (const float*)d_in[1];
  const float* elev         = (const float*)d_in[2];
  const int*   poses_idx    = (const int*)d_in[3];
  const int*   patch_idx    = (const int*)d_in[4];
  const float* target       = (const float*)d_in[5];
  const float* weights      = (const float*)d_in[6];
  float* out = (float*)d_out;

  const int n_obs       = in_sizes[3];            // 2,097,152
  const int pose_dwords = in_sizes[0];            // 4096*7 = 28672 (112 KB)
  const size_t sh_bytes = (size_t)pose_dwords * sizeof(float);

  // Allow >64 KB dynamic LDS (CDNA5 WGP has 320 KB). Idempotent/deterministic.
  (void)hipFuncSetAttribute((const void*)ba_residual_kernel,
                            hipFuncAttributeMaxDynamicSharedMemorySize,
                            (int)sh_bytes);

  const int grid = (n_obs + OBS_PER_BLK - 1) / OBS_PER_BLK;  // 1024 blocks
  ba_residual_kernel<<<grid, BLOCK_C, sh_bytes, stream>>>(
      poses, patch_coords, elev, poses_idx, patch_idx, target, weights, out,
      n_obs, pose_dwords);
}